// SelectiveScan2D_7567732375827
// MI455X (gfx1250) — compile-verified
//
#include <hip/hip_runtime.h>
#include <hip/hip_bf16.h>
#include <math.h>

// ---------------- types ----------------
typedef float v2f __attribute__((ext_vector_type(2)));
typedef float v8f __attribute__((ext_vector_type(8)));
typedef unsigned int v4u __attribute__((ext_vector_type(4)));
typedef int v4i __attribute__((ext_vector_type(4)));
typedef int v8i __attribute__((ext_vector_type(8)));

// ---------------- problem constants ----------------
#define M_TOT   16384      // 4*64*64 spatial positions
#define DIM_C   96
#define DI      192        // D_INNER
#define NS      16         // D_STATE
#define NPROJ   608        // 2*DI + 2*NS + DI
#define SEQS    256        // B*H == B*W
#define LSEQ    64

__device__ __forceinline__ float silu_f(float v) {
    return v * (1.0f / (1.0f + __expf(-v)));
}
__device__ __forceinline__ float softplus_f(float v) {
    return (v > 20.0f) ? v : log1pf(__expf(v));
}

// ---------------------------------------------------------------------------
// TDM: DMA a 2D tile (tile_d0 x tile_d1 elements of 4 bytes) from global
// memory into LDS at byte offset lds_off. Rows are stride0 elements apart in
// global memory; LDS destination is packed contiguously.
// D# layout per CDNA5 ISA ch. 8 (group0: count/lds/global/type, group1:
// data_size + dims + strides). Groups 2/3 zero (tile dims 2..4 unused -> 2D).
// Tracked by TENSORcnt.
// ---------------------------------------------------------------------------
__device__ __forceinline__ void tdm_load_2d_f32(unsigned lds_off,
                                                const void* gptr,
                                                unsigned tile_d0, unsigned tile_d1,
                                                unsigned tens_d0, unsigned tens_d1,
                                                unsigned long long stride0_elems)
{
    const unsigned long long ga = (unsigned long long)gptr;

    v4u g0;
    g0.x = 1u;                                            // count=1, user descriptor
    g0.y = lds_off;                                       // lds_addr (bytes)
    g0.z = (unsigned)(ga & 0xFFFFFFFFu);                  // global_addr[31:0]
    g0.w = (unsigned)((ga >> 32) & 0x01FFFFFFu)           // global_addr[56:32]
         | (2u << 30);                                    // type=2 ("image")

    const unsigned ds = 2u;                               // data_size: 2 -> 4 bytes
    v8i g1;
    g1[0] = (int)(ds << 16);                              // wg_mask=0, data_size
    g1[1] = (int)((tens_d0 & 0xFFFFu) << 16);             // tensor_dim0[15:0] @63:48
    g1[2] = (int)((tens_d0 >> 16) | ((tens_d1 & 0xFFFFu) << 16)); // d0 hi | d1 lo
    g1[3] = (int)((tens_d1 >> 16) | (tile_d0 << 16));     // d1 hi | tile_dim0
    g1[4] = (int)(tile_d1 & 0xFFFFu);                     // tile_dim1 | tile_dim2=0
    g1[5] = (int)(stride0_elems & 0xFFFFFFFFull);         // tensor_dim0_stride lo
    g1[6] = (int)((stride0_elems >> 32) & 0xFFFFull);     // stride0 hi | stride1 lo=0
    g1[7] = 0;                                            // stride1 hi = 0

    const v4i z4 = {0, 0, 0, 0};
#if __clang_major__ >= 23
    const v8i z8 = {0, 0, 0, 0, 0, 0, 0, 0};
    __builtin_amdgcn_tensor_load_to_lds(g0, g1, z4, z4, z8, 0);
#else
    __builtin_amdgcn_tensor_load_to_lds(g0, g1, z4, z4, 0);
#endif
}

// ---------------------------------------------------------------------------
// One wave computes a 16x16 FP32 tile of  A[M,K] @ Bt[N,K]^T  via
// V_WMMA_F32_16X16X4_F32 (ISA 7.12.2 fragment layouts).
// ---------------------------------------------------------------------------
__device__ __forceinline__ v8f wmma_tile_f32(const float* __restrict__ A, int lda,
                                             const float* __restrict__ Bt, int ldb,
                                             int m0, int n0, int K)
{
    const int lane = threadIdx.x & 31;
    const int half = lane >> 4;   // 0 or 1
    const int l16  = lane & 15;
    v8f acc = {0.f, 0.f, 0.f, 0.f, 0.f, 0.f, 0.f, 0.f};
    const float* __restrict__ arow = A  + (size_t)(m0 + l16) * lda;
    const float* __restrict__ brow = Bt + (size_t)(n0 + l16) * ldb;
    for (int k0 = 0; k0 < K; k0 += 4) {
        const int kk = k0 + 2 * half;
        v2f a = { arow[kk], arow[kk + 1] };
        v2f b = { brow[kk], brow[kk + 1] };
        acc = __builtin_amdgcn_wmma_f32_16x16x4_f32(
            /*neg_a=*/false, a, /*neg_b=*/false, b,
            /*c_mod=*/(short)0, acc, /*reuse_a=*/false, /*reuse_b=*/false);
    }
    return acc;
}

// ---------------------------------------------------------------------------
// Kernel 1: proj = x[M,96] @ W_in[608,96]^T, fused split + activations.
// ---------------------------------------------------------------------------
__global__ void ss2d_proj_kernel(const float* __restrict__ x,
                                 const float* __restrict__ W_in,
                                 const float* __restrict__ w_conv,
                                 float* __restrict__ xconv,
                                 float* __restrict__ zsilu,
                                 float* __restrict__ Bbuf,
                                 float* __restrict__ Cbuf,
                                 float* __restrict__ dbuf)
{
    const int NT   = NPROJ / 16;                       // 38
    const int wave = threadIdx.x >> 5;
    const int tile = blockIdx.x * (blockDim.x >> 5) + wave;
    const int mt = tile / NT, nt = tile % NT;
    const int m0 = mt * 16, n0 = nt * 16;

    v8f acc = wmma_tile_f32(x, DIM_C, W_in, DIM_C, m0, n0, DIM_C);

    const int lane = threadIdx.x & 31;
    const int half = lane >> 4;
    const int c    = n0 + (lane & 15);                 // global proj column

    if (n0 < DI) {
        const float wc = w_conv[c * 3 + 1];            // center conv tap
        #pragma unroll
        for (int v = 0; v < 8; ++v) {
            const int r = m0 + v + 8 * half;
            xconv[r * DI + c] = silu_f(acc[v] * wc);
        }
    } else if (n0 < 2 * DI) {
        const int cc = c - DI;
        #pragma unroll
        for (int v = 0; v < 8; ++v) {
            const int r = m0 + v + 8 * half;
            zsilu[r * DI + cc] = silu_f(acc[v]);
        }
    } else if (n0 < 2 * DI + NS) {
        const int cc = c - 2 * DI;
        #pragma unroll
        for (int v = 0; v < 8; ++v) {
            const int r = m0 + v + 8 * half;
            Bbuf[r * NS + cc] = acc[v];
        }
    } else if (n0 < 2 * DI + 2 * NS) {
        const int cc = c - (2 * DI + NS);
        #pragma unroll
        for (int v = 0; v < 8; ++v) {
            const int r = m0 + v + 8 * half;
            Cbuf[r * NS + cc] = acc[v];
        }
    } else {
        const int cc = c - (2 * DI + 2 * NS);
        #pragma unroll
        for (int v = 0; v < 8; ++v) {
            const int r = m0 + v + 8 * half;
            dbuf[r * DI + cc] = softplus_f(acc[v]);
        }
    }
}

// ---------------------------------------------------------------------------
// Kernel 2: selective scan. blockIdx.y = direction (0 = rows, 1 = columns),
// blockIdx.x = sequence, threadIdx.x = channel d. B/C for the whole sequence
// are DMA'd into LDS by the Tensor Data Mover (wave 0), recurrence state
// h[16] lives in registers, exp via native v_exp_f32.
// ---------------------------------------------------------------------------
__device__ __forceinline__ int seq_pos(int dir, int s, int l) {
    // dir 0: p = s*64 + l        (s = b*64+h, scan over w)
    // dir 1: p = (s/64)*4096 + l*64 + (s%64)  (s = b*64+w, scan over h)
    return dir ? (((s >> 6) << 12) + (l << 6) + (s & 63)) : ((s << 6) + l);
}

__global__ void ss2d_scan_kernel(const float* __restrict__ xconv,
                                 const float* __restrict__ dbuf,
                                 const float* __restrict__ Bbuf,
                                 const float* __restrict__ Cbuf,
                                 const float* __restrict__ A_log,
                                 const float* __restrict__ Dvec,
                                 float* __restrict__ ycat)
{
    const int s   = blockIdx.x;
    const int dir = blockIdx.y;
    const int d   = threadIdx.x;         // 0..191

    __shared__ float Bsh[LSEQ * NS];     // 4 KB
    __shared__ float Csh[LSEQ * NS];     // 4 KB

    // Wave 0 issues two TDM tile loads: 64 rows x 16 floats, row stride
    // 16 floats (horizontal) or 1024 floats (vertical). LDS dest is packed.
    if (threadIdx.x < 32) {
        const int p0 = seq_pos(dir, s, 0);
        const unsigned long long stride0 = dir ? (unsigned long long)(LSEQ * NS)
                                               : (unsigned long long)NS;
        const unsigned ldsB = (unsigned)(size_t)(void*)Bsh;
        const unsigned ldsC = (unsigned)(size_t)(void*)Csh;
        tdm_load_2d_f32(ldsB, Bbuf + (size_t)p0 * NS, NS, LSEQ, NS, LSEQ, stride0);
        tdm_load_2d_f32(ldsC, Cbuf + (size_t)p0 * NS, NS, LSEQ, NS, LSEQ, stride0);
        __builtin_amdgcn_s_wait_tensorcnt(0);
    }
    __syncthreads();

    float Arow[NS];
    #pragma unroll
    for (int n = 0; n < NS; ++n)
        Arow[n] = -__expf(A_log[d * NS + n]);          // A = -exp(A_log)
    const float Dd = Dvec[d];

    float h[NS];
    #pragma unroll
    for (int n = 0; n < NS; ++n) h[n] = 0.f;

    for (int l = 0; l < LSEQ; ++l) {
        const int   p  = seq_pos(dir, s, l);
        const float xv = xconv[p * DI + d];
        const float dv = dbuf[p * DI + d];
        float y = 0.f;
        #pragma unroll
        for (int n = 0; n < NS; ++n) {
            const float dA  = __expf(dv * Arow[n]);
            const float hn  = dA * h[n] + dv * Bsh[l * NS + n] * xv;
            h[n] = hn;
            y += hn * Csh[l * NS + n];
        }
        y += xv * Dd;
        ycat[p * (2 * DI) + dir * DI + d] = y;
    }
}

// ---------------------------------------------------------------------------
// Kernel 3: yf = (ycat[M,384] @ W_fuse[192,384]^T) * zsilu
// ---------------------------------------------------------------------------
__global__ void ss2d_fuse_kernel(const float* __restrict__ ycat,
                                 const float* __restrict__ W_fuse,
                                 const float* __restrict__ zsilu,
                                 float* __restrict__ yf)
{
    const int NT   = DI / 16;                          // 12
    const int wave = threadIdx.x >> 5;
    const int tile = blockIdx.x * (blockDim.x >> 5) + wave;
    const int m0 = (tile / NT) * 16, n0 = (tile % NT) * 16;

    v8f acc = wmma_tile_f32(ycat, 2 * DI, W_fuse, 2 * DI, m0, n0, 2 * DI);

    const int lane = threadIdx.x & 31;
    const int half = lane >> 4;
    const int c    = n0 + (lane & 15);
    #pragma unroll
    for (int v = 0; v < 8; ++v) {
        const int r = m0 + v + 8 * half;
        yf[r * DI + c] = acc[v] * zsilu[r * DI + c];
    }
}

// ---------------------------------------------------------------------------
// Kernel 4: out = yf[M,192] @ W_out[96,192]^T
// ---------------------------------------------------------------------------
__global__ void ss2d_out_kernel(const float* __restrict__ yf,
                                const float* __restrict__ W_out,
                                float* __restrict__ out)
{
    const int NT   = DIM_C / 16;                       // 6
    const int wave = threadIdx.x >> 5;
    const int tile = blockIdx.x * (blockDim.x >> 5) + wave;
    const int m0 = (tile / NT) * 16, n0 = (tile % NT) * 16;

    v8f acc = wmma_tile_f32(yf, DI, W_out, DI, m0, n0, DI);

    const int lane = threadIdx.x & 31;
    const int half = lane >> 4;
    const int c    = n0 + (lane & 15);
    #pragma unroll
    for (int v = 0; v < 8; ++v) {
        const int r = m0 + v + 8 * half;
        out[r * DIM_C + c] = acc[v];
    }
}

// ---------------------------------------------------------------------------
extern "C" void kernel_launch(void* const* d_in, const int* in_sizes, int n_in,
                              void* d_out, int out_size, void* d_ws, size_t ws_size,
                              hipStream_t stream)
{
    const float* x      = (const float*)d_in[0];   // [4,64,64,96]
    const float* W_in   = (const float*)d_in[1];   // [608,96]
    const float* w_conv = (const float*)d_in[2];   // [192,1,3]
    const float* W_fuse = (const float*)d_in[3];   // [192,384,1]
    const float* W_out  = (const float*)d_in[4];   // [96,192]
    const float* A_log  = (const float*)d_in[5];   // [192,16]
    const float* Dvec   = (const float*)d_in[6];   // [192]
    float* out          = (float*)d_out;           // [16384,96]

    // workspace partition (all fp32)
    char* ws = (char*)d_ws;
    size_t off = 0;
    float* xconv = (float*)(ws + off); off += (size_t)M_TOT * DI * 4;      // 12.6 MB
    float* zsilu = (float*)(ws + off); off += (size_t)M_TOT * DI * 4;      // 12.6 MB
    float* dbuf  = (float*)(ws + off); off += (size_t)M_TOT * DI * 4;      // 12.6 MB
    float* Bbuf  = (float*)(ws + off); off += (size_t)M_TOT * NS * 4;      //  1.0 MB
    float* Cbuf  = (float*)(ws + off); off += (size_t)M_TOT * NS * 4;      //  1.0 MB
    float* ycat  = (float*)(ws + off); off += (size_t)M_TOT * 2 * DI * 4;  // 25.2 MB
    float* yf    = (float*)(ws + off); off += (size_t)M_TOT * DI * 4;      // 12.6 MB

    // 1) projection GEMM + split/activation epilogue
    {
        const int tiles = (M_TOT / 16) * (NPROJ / 16);  // 38912
        ss2d_proj_kernel<<<tiles / 8, 256, 0, stream>>>(
            x, W_in, w_conv, xconv, zsilu, Bbuf, Cbuf, dbuf);
    }
    // 2) horizontal + vertical selective scans (TDM-staged B/C)
    {
        dim3 grid(SEQS, 2);
        ss2d_scan_kernel<<<grid, DI, 0, stream>>>(
            xconv, dbuf, Bbuf, Cbuf, A_log, Dvec, ycat);
    }
    // 3) fuse GEMM * silu(z)
    {
        const int tiles = (M_TOT / 16) * (DI / 16);     // 12288
        ss2d_fuse_kernel<<<tiles / 8, 256, 0, stream>>>(ycat, W_fuse, zsilu, yf);
    }
    // 4) output GEMM
    {
        const int tiles = (M_TOT / 16) * (DIM_C / 16);  // 6144
        ss2d_out_kernel<<<tiles / 8, 256, 0, stream>>>(yf, W_out, out);
    }
}